// SpookyLocalBlock_47064251630056
// MI455X (gfx1250) — compile-verified
//
#include <hip/hip_runtime.h>

// ---------------------------------------------------------------------------
// CDNA5 (gfx1250) implementation using V_WMMA_F32_16X16X4_F32 (wave32).
//
// WMMA operand layout assumptions (from CDNA5 ISA 7.12.2):
//   A (16x4 f32, 2 VGPR):  lane L holds row m=L&15; v[0]=A[m][k0+2*half],
//                          v[1]=A[m][k0+2*half+1]  (half = L>>4)
//   B (4x16 f32, 2 VGPR):  lane L holds col n=L&15; v[0]=B[k0+2*half][n], v[1]=...+1
//   C/D (16x16 f32, 8 VGPR): reg r, lane L -> D[r + 8*half][L&15]
// ---------------------------------------------------------------------------

typedef __attribute__((ext_vector_type(2))) float v2f;
typedef __attribute__((ext_vector_type(8))) float v8f;

#define FDIM 128
#define XS   132   // LDS row stride for 128-wide tiles (conflict-free, 8B-aligned)
#define AS   36    // LDS row stride for K=32 A-tiles  (conflict-free, 8B-aligned)

__device__ __forceinline__ v8f wmma4(v2f a, v2f b, v8f c) {
  // D = A(16x4) * B(4x16) + C, fp32
  return __builtin_amdgcn_wmma_f32_16x16x4_f32(false, a, false, b,
                                               (short)0, c, false, false);
}

__device__ __forceinline__ float silu(float x) {
  return x / (1.0f + __expf(-x));
}

// ---------------------------------------------------------------------------
// Kernel A: one ResMLP channel.  out = silu(h) @ W1^T + b1,
//           h = x + silu(x) @ Wr^T + br.
// Block: 256 threads (8 waves), 16 atoms x 128 features.
// ---------------------------------------------------------------------------
__global__ __launch_bounds__(256) void resmlp_kernel(
    const float* __restrict__ X,
    const float* __restrict__ Wr, const float* __restrict__ Br,
    const float* __restrict__ W1, const float* __restrict__ B1,
    float* __restrict__ Dst)
{
  __shared__ float xs[16 * XS];   // raw x tile
  __shared__ float ss[16 * XS];   // silu(x), later silu(h)

  const int tid = threadIdx.x;
  const int ab  = blockIdx.x * 16;

  for (int i = tid; i < 16 * FDIM; i += 256) {
    int m = i >> 7, k = i & 127;
    float v = X[(size_t)(ab + m) * FDIM + k];
    xs[m * XS + k] = v;
    ss[m * XS + k] = silu(v);
  }
  __syncthreads();

  const int lane = tid & 31;
  const int wave = tid >> 5;
  const int n    = lane & 15;
  const int half = lane >> 4;
  const int f    = wave * 16 + n;
  const int kk   = half * 2;

  const float brf = Br[f];
  const float b1f = B1[f];

  // GEMM1: acc[m][f] = sum_k silu(x)[m][k] * Wr[f][k]
  v8f acc = {0.f,0.f,0.f,0.f,0.f,0.f,0.f,0.f};
#pragma unroll
  for (int k0 = 0; k0 < FDIM; k0 += 4) {
    v2f a = *(const v2f*)&ss[n * XS + k0 + kk];
    v2f b = *(const v2f*)&Wr[(size_t)f * FDIM + k0 + kk];
    acc = wmma4(a, b, acc);
  }
  __syncthreads();   // all waves done reading silu(x) before overwrite

  // h = x + acc + br ; stage silu(h) into LDS (block transpose)
#pragma unroll
  for (int r = 0; r < 8; ++r) {
    int m = r + 8 * half;
    float h = xs[m * XS + f] + acc[r] + brf;
    ss[m * XS + f] = silu(h);
  }
  __syncthreads();

  // GEMM2: out[m][f] = sum_k silu(h)[m][k] * W1[f][k] + b1
  v8f acc2 = {0.f,0.f,0.f,0.f,0.f,0.f,0.f,0.f};
#pragma unroll
  for (int k0 = 0; k0 < FDIM; k0 += 4) {
    v2f a = *(const v2f*)&ss[n * XS + k0 + kk];
    v2f b = *(const v2f*)&W1[(size_t)f * FDIM + k0 + kk];
    acc2 = wmma4(a, b, acc2);
  }
#pragma unroll
  for (int r = 0; r < 8; ++r) {
    int m = r + 8 * half;
    Dst[(size_t)(ab + m) * FDIM + f] = acc2[r] + b1f;
  }
}

// ---------------------------------------------------------------------------
// Kernel B: env projections + neighbor gather-reduce + quadratic heads.
// Block: 256 threads (8 waves), 4 atoms.
//   s:  env_s = gs @ Gs^T (per atom: 16x32 * 32x128), final_s = sum_j sfea*env
//   p/d: all_p[f][a] staged to LDS, then (Vp@P1)·(Vp@P2) row-dots -> scalar
// ---------------------------------------------------------------------------
__global__ __launch_bounds__(256) void env_kernel(
    const int*   __restrict__ Idx,
    const float* __restrict__ Gs_g, const float* __restrict__ Gp_g,
    const float* __restrict__ Gd_g,
    const float* __restrict__ P1,   const float* __restrict__ P2,
    const float* __restrict__ D1,   const float* __restrict__ D2,
    const float* __restrict__ Sf,   const float* __restrict__ Pf,
    const float* __restrict__ Df,
    const float* __restrict__ gs,   const float* __restrict__ gp,
    const float* __restrict__ gd,
    float* __restrict__ Out)
{
  // A-tile rows: s: [0,64), p: [64,256), d: [256,576)   (16 rows per tile)
  __shared__ float lA[576 * AS];      // ~81 KB
  __shared__ float lVp[16 * XS];      // all_p rows (atom*3+a), padded w/ zeros
  __shared__ float lVd[32 * XS];      // all_d rows (atom*5+a), padded w/ zeros
  __shared__ float lFP[4], lFD[4];
  __shared__ int   lIdx[48];

  const int tid = threadIdx.x;
  const int ab  = blockIdx.x * 4;

  // ---- zero pass -----------------------------------------------------------
  for (int i = tid; i < 576 * AS; i += 256) lA[i]  = 0.f;
  for (int i = tid; i < 16 * XS;  i += 256) lVp[i] = 0.f;
  for (int i = tid; i < 32 * XS;  i += 256) lVd[i] = 0.f;
  if (tid < 4) { lFP[tid] = 0.f; lFD[tid] = 0.f; }
  if (tid < 48) lIdx[tid] = Idx[(size_t)ab * 12 + tid];
  __syncthreads();

  // ---- stage gs/gp/gd into zero-padded WMMA A-tiles ------------------------
  for (int i = tid; i < 4 * 12 * 32; i += 256) {          // s: contiguous
    int atom = i / 384, j = (i / 32) % 12, k = i & 31;
    lA[(atom * 16 + j) * AS + k] = gs[(size_t)ab * 384 + i];
  }
  for (int i = tid; i < 4 * 12 * 32 * 3; i += 256) {      // p: a innermost
    int atom = i / 1152, r = i % 1152;
    int j = r / 96, k = (r / 3) % 32, a = r % 3;
    lA[(64 + (atom * 3 + a) * 16 + j) * AS + k] = gp[(size_t)ab * 1152 + i];
  }
  for (int i = tid; i < 4 * 12 * 32 * 5; i += 256) {      // d: a innermost
    int atom = i / 1920, r = i % 1920;
    int j = r / 160, k = (r / 5) % 32, a = r % 5;
    lA[(256 + (atom * 5 + a) * 16 + j) * AS + k] = gd[(size_t)ab * 1920 + i];
  }
  __syncthreads();

  const int lane = tid & 31;
  const int wave = tid >> 5;
  const int n    = lane & 15;
  const int half = lane >> 4;
  const int f    = wave * 16 + n;   // this wave's feature column
  const int kk   = half * 2;

  // preload B operands (G[f][k] columns) for the K=32 env GEMMs
  v2f Bs[8], Bp[8], Bd[8];
#pragma unroll
  for (int t = 0; t < 8; ++t) {
    Bs[t] = *(const v2f*)&Gs_g[(size_t)f * 32 + t * 4 + kk];
    Bp[t] = *(const v2f*)&Gp_g[(size_t)f * 32 + t * 4 + kk];
    Bd[t] = *(const v2f*)&Gd_g[(size_t)f * 32 + t * 4 + kk];
  }

  float sRes[4];

#pragma unroll
  for (int atom = 0; atom < 4; ++atom) {
    // gather neighbor features for this lane's rows (m = r + 8*half)
    float gS[8], gP[8], gD[8];
#pragma unroll
    for (int r = 0; r < 8; ++r) {
      int m = r + 8 * half;
      if (m < 12) {
        int nb = lIdx[atom * 12 + m];
        gS[r] = Sf[(size_t)nb * FDIM + f];
        gP[r] = Pf[(size_t)nb * FDIM + f];
        gD[r] = Df[(size_t)nb * FDIM + f];
      } else { gS[r] = 0.f; gP[r] = 0.f; gD[r] = 0.f; }
    }

    // ---- s channel ----
    {
      const float* Ar = &lA[(atom * 16 + n) * AS];
      v8f acc = {0.f,0.f,0.f,0.f,0.f,0.f,0.f,0.f};
#pragma unroll
      for (int t = 0; t < 8; ++t)
        acc = wmma4(*(const v2f*)&Ar[t * 4 + kk], Bs[t], acc);
      float part = 0.f;
#pragma unroll
      for (int r = 0; r < 8; ++r) part += gS[r] * acc[r];
      part += __shfl_xor(part, 16, 32);
      sRes[atom] = part;
    }

    // ---- p channel: 3 components -> rows of Vp ----
#pragma unroll
    for (int a = 0; a < 3; ++a) {
      const float* Ar = &lA[(64 + (atom * 3 + a) * 16 + n) * AS];
      v8f acc = {0.f,0.f,0.f,0.f,0.f,0.f,0.f,0.f};
#pragma unroll
      for (int t = 0; t < 8; ++t)
        acc = wmma4(*(const v2f*)&Ar[t * 4 + kk], Bp[t], acc);
      float part = 0.f;
#pragma unroll
      for (int r = 0; r < 8; ++r) part += gP[r] * acc[r];
      part += __shfl_xor(part, 16, 32);
      if (half == 0) lVp[(atom * 3 + a) * XS + f] = part;
    }

    // ---- d channel: 5 components -> rows of Vd ----
#pragma unroll
    for (int a = 0; a < 5; ++a) {
      const float* Ar = &lA[(256 + (atom * 5 + a) * 16 + n) * AS];
      v8f acc = {0.f,0.f,0.f,0.f,0.f,0.f,0.f,0.f};
#pragma unroll
      for (int t = 0; t < 8; ++t)
        acc = wmma4(*(const v2f*)&Ar[t * 4 + kk], Bd[t], acc);
      float part = 0.f;
#pragma unroll
      for (int r = 0; r < 8; ++r) part += gD[r] * acc[r];
      part += __shfl_xor(part, 16, 32);
      if (half == 0) lVd[(atom * 5 + a) * XS + f] = part;
    }
  }
  __syncthreads();

  // ---- quadratic heads: sum_a (v_a P1)·(v_a P2) ---------------------------
  // p channel: one 16-row M-tile (rows atom*3+a, 12 valid, rest zero)
  {
    v8f q = {0.f,0.f,0.f,0.f,0.f,0.f,0.f,0.f};
    v8f w = {0.f,0.f,0.f,0.f,0.f,0.f,0.f,0.f};
#pragma unroll
    for (int k0 = 0; k0 < FDIM; k0 += 4) {
      v2f a = *(const v2f*)&lVp[n * XS + k0 + kk];
      v2f b1 = { P1[(size_t)(k0 + kk) * FDIM + f], P1[(size_t)(k0 + kk + 1) * FDIM + f] };
      v2f b2 = { P2[(size_t)(k0 + kk) * FDIM + f], P2[(size_t)(k0 + kk + 1) * FDIM + f] };
      q = wmma4(a, b1, q);
      w = wmma4(a, b2, w);
    }
#pragma unroll
    for (int r = 0; r < 8; ++r) {
      float pr = q[r] * w[r];
      pr += __shfl_xor(pr, 1, 32);
      pr += __shfl_xor(pr, 2, 32);
      pr += __shfl_xor(pr, 4, 32);
      pr += __shfl_xor(pr, 8, 32);
      if (n == 0) {
        int m = r + 8 * half;
        if (m < 12) atomicAdd(&lFP[m / 3], pr);
      }
    }
  }
  // d channel: two 16-row M-tiles (rows atom*5+a, 20 valid)
#pragma unroll
  for (int mt = 0; mt < 2; ++mt) {
    v8f q = {0.f,0.f,0.f,0.f,0.f,0.f,0.f,0.f};
    v8f w = {0.f,0.f,0.f,0.f,0.f,0.f,0.f,0.f};
#pragma unroll
    for (int k0 = 0; k0 < FDIM; k0 += 4) {
      v2f a = *(const v2f*)&lVd[(mt * 16 + n) * XS + k0 + kk];
      v2f b1 = { D1[(size_t)(k0 + kk) * FDIM + f], D1[(size_t)(k0 + kk + 1) * FDIM + f] };
      v2f b2 = { D2[(size_t)(k0 + kk) * FDIM + f], D2[(size_t)(k0 + kk + 1) * FDIM + f] };
      q = wmma4(a, b1, q);
      w = wmma4(a, b2, w);
    }
#pragma unroll
    for (int r = 0; r < 8; ++r) {
      float pr = q[r] * w[r];
      pr += __shfl_xor(pr, 1, 32);
      pr += __shfl_xor(pr, 2, 32);
      pr += __shfl_xor(pr, 4, 32);
      pr += __shfl_xor(pr, 8, 32);
      if (n == 0) {
        int m = mt * 16 + r + 8 * half;
        if (m < 20) atomicAdd(&lFD[m / 5], pr);
      }
    }
  }
  __syncthreads();

  // ---- final accumulation into output (out already holds c-channel) -------
  if (half == 0) {
#pragma unroll
    for (int atom = 0; atom < 4; ++atom) {
      size_t o = (size_t)(ab + atom) * FDIM + f;
      Out[o] += sRes[atom] + lFP[atom] + lFD[atom];
    }
  }
}

// ---------------------------------------------------------------------------
extern "C" void kernel_launch(void* const* d_in, const int* in_sizes, int n_in,
                              void* d_out, int out_size, void* d_ws, size_t ws_size,
                              hipStream_t stream) {
  const float* atom_fea = (const float*)d_in[0];
  const int*   nbr      = (const int*)  d_in[1];
  const float* gs       = (const float*)d_in[2];
  const float* gp       = (const float*)d_in[3];
  const float* gd       = (const float*)d_in[4];
  const float* Gs       = (const float*)d_in[5];
  const float* Gp       = (const float*)d_in[6];
  const float* Gd       = (const float*)d_in[7];
  const float* P1       = (const float*)d_in[8];
  const float* P2       = (const float*)d_in[9];
  const float* D1       = (const float*)d_in[10];
  const float* D2       = (const float*)d_in[11];
  const float* c_wr = (const float*)d_in[12];
  const float* c_br = (const float*)d_in[13];
  const float* c_w1 = (const float*)d_in[14];
  const float* c_b1 = (const float*)d_in[15];
  const float* s_wr = (const float*)d_in[16];
  const float* s_br = (const float*)d_in[17];
  const float* s_w1 = (const float*)d_in[18];
  const float* s_b1 = (const float*)d_in[19];
  const float* p_wr = (const float*)d_in[20];
  const float* p_br = (const float*)d_in[21];
  const float* p_w1 = (const float*)d_in[22];
  const float* p_b1 = (const float*)d_in[23];
  const float* d_wr = (const float*)d_in[24];
  const float* d_br = (const float*)d_in[25];
  const float* d_w1 = (const float*)d_in[26];
  const float* d_b1 = (const float*)d_in[27];

  float* out  = (float*)d_out;
  const int N = in_sizes[0] / FDIM;   // 30000

  float* sfea = (float*)d_ws;
  float* pfea = sfea + (size_t)N * FDIM;
  float* dfea = pfea + (size_t)N * FDIM;

  dim3 blk(256);
  // c channel writes d_out directly; s/p/d go to workspace
  resmlp_kernel<<<N / 16, blk, 0, stream>>>(atom_fea, c_wr, c_br, c_w1, c_b1, out);
  resmlp_kernel<<<N / 16, blk, 0, stream>>>(atom_fea, s_wr, s_br, s_w1, s_b1, sfea);
  resmlp_kernel<<<N / 16, blk, 0, stream>>>(atom_fea, p_wr, p_br, p_w1, p_b1, pfea);
  resmlp_kernel<<<N / 16, blk, 0, stream>>>(atom_fea, d_wr, d_br, d_w1, d_b1, dfea);

  env_kernel<<<N / 4, blk, 0, stream>>>(nbr, Gs, Gp, Gd, P1, P2, D1, D2,
                                        sfea, pfea, dfea, gs, gp, gd, out);
}